// KGINModel_40106404610266
// MI455X (gfx1250) — compile-verified
//
#include <hip/hip_runtime.h>
#include <math.h>

// ---------------- problem constants ----------------
#define N_USERS 100000
#define N_ENT   200000
#define CH      64
#define NFAC    4
#define NRELM1  9          // N_RELATIONS - 1
#define NEDGE   2000000
#define NNZV    1000000
#define NHOPS   2

// ---------------- workspace layout (float offsets) ----------------
#define OFF_ENT_CUR 0L            // 12,800,000
#define OFF_USR_CUR 12800000L     //  6,400,000
#define OFF_ENT_AGG 19200000L     // 12,800,000
#define OFF_USR_AGG 32000000L     //  6,400,000
#define OFF_CNT     38400000L     //    200,000
#define OFF_NORM    38600000L     //          2   [0]=entity sumsq, [1]=user sumsq
#define OFF_DW      38600004L     //        256   disen_w = softmax(att) @ weight, [F][C]
// total ~154.4 MB of workspace

typedef __attribute__((ext_vector_type(2))) float v2f;
typedef __attribute__((ext_vector_type(8))) float v8f;

// ---------------------------------------------------------------
// cor + disen_w (tiny, one block)
// ---------------------------------------------------------------
__global__ void k_small(const float* __restrict__ att, const float* __restrict__ weight,
                        float* __restrict__ dw, float* __restrict__ cor_out) {
  __shared__ float sm[NFAC * NRELM1];
  int tid = threadIdx.x;
  if (tid == 0) {
    // cosine correlation over factor pairs
    float nrm[NFAC];
    for (int f = 0; f < NFAC; ++f) {
      float s = 0.f;
      for (int r = 0; r < NRELM1; ++r) { float v = att[f*NRELM1+r]; s += v*v; }
      nrm[f] = sqrtf(s);
    }
    float cor = 0.f;
    for (int i = 0; i < NFAC; ++i)
      for (int j = i+1; j < NFAC; ++j) {
        float d = 0.f;
        for (int r = 0; r < NRELM1; ++r) d += att[i*NRELM1+r]*att[j*NRELM1+r];
        d /= (nrm[i]*nrm[j]);
        cor += d*d;
      }
    *cor_out = cor;
    // row-wise softmax of att into LDS
    for (int f = 0; f < NFAC; ++f) {
      float mx = -3.0e38f;
      for (int r = 0; r < NRELM1; ++r) mx = fmaxf(mx, att[f*NRELM1+r]);
      float s = 0.f;
      for (int r = 0; r < NRELM1; ++r) { float e = expf(att[f*NRELM1+r]-mx); sm[f*NRELM1+r]=e; s += e; }
      float inv = 1.0f/s;
      for (int r = 0; r < NRELM1; ++r) sm[f*NRELM1+r] *= inv;
    }
  }
  __syncthreads();
  if (tid < CH) {
    for (int f = 0; f < NFAC; ++f) {
      float acc = 0.f;
      for (int r = 0; r < NRELM1; ++r) acc += sm[f*NRELM1+r]*weight[r*CH+tid];
      dw[f*CH+tid] = acc;
    }
  }
}

// ---------------------------------------------------------------
// init: cur = emb, res = emb  (float4, 4.8M vectors)
// ---------------------------------------------------------------
__global__ void k_init(const float* __restrict__ user_emb, const float* __restrict__ entity_emb,
                       float* __restrict__ ws, float* __restrict__ out) {
  long i4 = (long)blockIdx.x*blockDim.x + threadIdx.x;
  const long NE4 = (long)N_ENT*CH/4, NU4 = (long)N_USERS*CH/4;
  if (i4 < NE4) {
    float4 v = ((const float4*)entity_emb)[i4];
    ((float4*)(ws + OFF_ENT_CUR))[i4] = v;
    ((float4*)out)[i4] = v;
  } else if (i4 < NE4 + NU4) {
    long j = i4 - NE4;
    float4 v = ((const float4*)user_emb)[j];
    ((float4*)(ws + OFF_USR_CUR))[j] = v;
    ((float4*)out)[NE4 + j] = v;
  }
}

// ---------------------------------------------------------------
// per-hop zero of agg buffers + cnt + norm accumulators
// ---------------------------------------------------------------
__global__ void k_zero(float* __restrict__ ws) {
  long i4 = (long)blockIdx.x*blockDim.x + threadIdx.x;
  const long Z4 = 19400000L/4;   // ent_agg + usr_agg + cnt, contiguous
  if (i4 < Z4) ((float4*)(ws + OFF_ENT_AGG))[i4] = make_float4(0.f,0.f,0.f,0.f);
  if (i4 == 0) { ws[OFF_NORM] = 0.f; ws[OFF_NORM+1] = 0.f; }
}

// ---------------------------------------------------------------
// KG edge scatter: ent_agg[head] += ent_cur[tail] * weight[type-1]; cnt[head] += 1
// 4 threads / edge, 16 channels (4x float4) each
// ---------------------------------------------------------------
__global__ void k_edge(const float* __restrict__ ent_cur, const float* __restrict__ weight,
                       const int* __restrict__ eh, const int* __restrict__ et,
                       const int* __restrict__ ety,
                       float* __restrict__ ent_agg, float* __restrict__ cnt) {
  long gid = (long)blockIdx.x*blockDim.x + threadIdx.x;
  if (gid >= (long)NEDGE*4) return;
  int e = (int)(gid >> 2), part = (int)(gid & 3);
  int head = eh[e], tail = et[e], t = ety[e] - 1;
  const float4* tv = (const float4*)(ent_cur + (long)tail*CH + part*16);
  const float4* wv = (const float4*)(weight  + (long)t   *CH + part*16);
  float* dst = ent_agg + (long)head*CH + part*16;
#pragma unroll
  for (int k = 0; k < 4; ++k) {
    float4 a = tv[k], b = wv[k];
    atomicAdd(dst + 4*k + 0, a.x*b.x);
    atomicAdd(dst + 4*k + 1, a.y*b.y);
    atomicAdd(dst + 4*k + 2, a.z*b.z);
    atomicAdd(dst + 4*k + 3, a.w*b.w);
  }
  if (part == 0) atomicAdd(cnt + head, 1.0f);
}

// ---------------------------------------------------------------
// user scatter: usr_agg[ui] += vals * ent_cur[ei]
// ---------------------------------------------------------------
__global__ void k_nnz(const float* __restrict__ ent_cur, const float* __restrict__ vals,
                      const int* __restrict__ ui, const int* __restrict__ ei,
                      float* __restrict__ usr_agg) {
  long gid = (long)blockIdx.x*blockDim.x + threadIdx.x;
  if (gid >= (long)NNZV*4) return;
  int nz = (int)(gid >> 2), part = (int)(gid & 3);
  float v = vals[nz];
  int u = ui[nz], e = ei[nz];
  const float4* sv = (const float4*)(ent_cur + (long)e*CH + part*16);
  float* dst = usr_agg + (long)u*CH + part*16;
#pragma unroll
  for (int k = 0; k < 4; ++k) {
    float4 a = sv[k];
    atomicAdd(dst + 4*k + 0, v*a.x);
    atomicAdd(dst + 4*k + 1, v*a.y);
    atomicAdd(dst + 4*k + 2, v*a.z);
    atomicAdd(dst + 4*k + 3, v*a.w);
  }
}

// ---------------------------------------------------------------
// entity mean + global sum-of-squares (block reduce -> 1 atomic/block)
// ---------------------------------------------------------------
__global__ void k_entdiv(float* __restrict__ ent_agg, const float* __restrict__ cnt,
                         float* __restrict__ norm) {
  long i4 = (long)blockIdx.x*blockDim.x + threadIdx.x;   // 3.2M exactly
  long row = i4 >> 4;                                     // 16 float4 per row
  float inv = 1.0f / fmaxf(cnt[row], 1.0f);
  float4 v = ((float4*)ent_agg)[i4];
  v.x *= inv; v.y *= inv; v.z *= inv; v.w *= inv;
  ((float4*)ent_agg)[i4] = v;
  float ss = v.x*v.x + v.y*v.y + v.z*v.z + v.w*v.w;
#pragma unroll
  for (int off = 16; off; off >>= 1) ss += __shfl_xor(ss, off, 32);
  __shared__ float red[8];
  int wv_ = threadIdx.x >> 5, ln = threadIdx.x & 31;
  if (ln == 0) red[wv_] = ss;
  __syncthreads();
  if (threadIdx.x == 0) {
    float s = 0.f;
#pragma unroll
    for (int w = 0; w < 8; ++w) s += red[w];
    atomicAdd(norm, s);
  }
}

// ---------------------------------------------------------------
// WMMA: per 16-user tile, P = U(16x64) @ latentT(64x4) via
// 16x v_wmma_f32_16x16x4_f32 (exact f32), then softmax over 4 factors,
// usr_agg *= (1 + score @ disen_w), accumulate user sum-of-squares.
// One wave32 per tile; 100000/16 = 6250 exact tiles -> EXEC all-ones.
// ---------------------------------------------------------------
__global__ void __launch_bounds__(32) k_userscore(
    const float* __restrict__ usr_cur, const float* __restrict__ latent,
    const float* __restrict__ dw_g, float* __restrict__ usr_agg,
    float* __restrict__ norm) {
  __shared__ __align__(16) float sdw[NFAC*CH];   // disen_w [F][C]
  __shared__ float sP[16*NFAC];                  // score logits tile
  int lane = threadIdx.x;
  long ubase = (long)blockIdx.x * 16;

  // stage disen_w into LDS (256 floats / 32 lanes)
#pragma unroll
  for (int k = 0; k < 8; ++k) sdw[lane + 32*k] = dw_g[lane + 32*k];

  // ---- WMMA accumulation over K=64 in steps of 4 ----
  // A 16x4 f32: lane L -> m=L&15; VGPR0 holds K=(hi?2:0), VGPR1 K=(hi?3:1)
  // B 4x16 f32: lane L -> n=L&15; VGPR0 row K=(hi?2:0), VGPR1 row K=(hi?3:1)
  int m  = lane & 15;
  int hi = lane >> 4;
  int n  = lane & 15;
  int nc = (n < NFAC) ? n : 0;
  float nmask = (n < NFAC) ? 1.0f : 0.0f;       // branch-free: keep EXEC full
  const float* urow = usr_cur + (ubase + m)*CH;
  const float* lrow = latent + (long)nc*CH;
  v8f acc = {0.f,0.f,0.f,0.f,0.f,0.f,0.f,0.f};
#pragma unroll
  for (int ks = 0; ks < 16; ++ks) {
    int kb = ks*4 + hi*2;
    v2f a; a.x = urow[kb];           a.y = urow[kb+1];
    v2f b; b.x = lrow[kb]*nmask;     b.y = lrow[kb+1]*nmask;
    acc = __builtin_amdgcn_wmma_f32_16x16x4_f32(
        false, a, false, b, (short)0, acc, false, false);
  }

  // D layout: VGPR r -> M = r + (hi?8:0), N = lane&15. Dump live columns to LDS.
#pragma unroll
  for (int r = 0; r < 8; ++r) {
    int mm = r + hi*8;
    if (n < NFAC) sP[mm*NFAC + n] = acc[r];
  }
  __syncthreads();

  // ---- phase 2: softmax(4) + modulate 64 channels; 2 lanes / user ----
  int ul   = lane >> 1;       // user within tile
  int half = lane & 1;        // channel half (32 each)
  float p0 = sP[ul*4+0], p1 = sP[ul*4+1], p2 = sP[ul*4+2], p3 = sP[ul*4+3];
  float mx = fmaxf(fmaxf(p0,p1), fmaxf(p2,p3));
  float e0 = expf(p0-mx), e1 = expf(p1-mx), e2 = expf(p2-mx), e3 = expf(p3-mx);
  float inv = 1.0f/(e0+e1+e2+e3);
  float s0 = e0*inv, s1 = e1*inv, s2 = e2*inv, s3 = e3*inv;

  float* row = usr_agg + (ubase + ul)*CH + half*32;
  const float4* w0 = (const float4*)(sdw + 0*CH + half*32);
  const float4* w1 = (const float4*)(sdw + 1*CH + half*32);
  const float4* w2 = (const float4*)(sdw + 2*CH + half*32);
  const float4* w3 = (const float4*)(sdw + 3*CH + half*32);
  float ss = 0.f;
#pragma unroll
  for (int c4 = 0; c4 < 8; ++c4) {
    float4 v = ((float4*)row)[c4];
    float4 a = w0[c4], b = w1[c4], c = w2[c4], d = w3[c4];
    float mx_ = 1.0f + s0*a.x + s1*b.x + s2*c.x + s3*d.x;
    float my_ = 1.0f + s0*a.y + s1*b.y + s2*c.y + s3*d.y;
    float mz_ = 1.0f + s0*a.z + s1*b.z + s2*c.z + s3*d.z;
    float mw_ = 1.0f + s0*a.w + s1*b.w + s2*c.w + s3*d.w;
    v.x *= mx_; v.y *= my_; v.z *= mz_; v.w *= mw_;
    ((float4*)row)[c4] = v;
    ss += v.x*v.x + v.y*v.y + v.z*v.z + v.w*v.w;
  }
#pragma unroll
  for (int off = 16; off; off >>= 1) ss += __shfl_xor(ss, off, 32);
  if (lane == 0) atomicAdd(norm + 1, ss);
}

// ---------------------------------------------------------------
// global-L2 normalize, update cur embeddings, accumulate residuals
// ---------------------------------------------------------------
__global__ void k_finalize(float* __restrict__ ws, float* __restrict__ out) {
  long i4 = (long)blockIdx.x*blockDim.x + threadIdx.x;   // 4.8M exactly
  const long NE4 = (long)N_ENT*CH/4, NU4 = (long)N_USERS*CH/4;
  if (i4 < NE4) {
    float rinv = 1.0f/sqrtf(ws[OFF_NORM]);
    float4 v = ((float4*)(ws + OFF_ENT_AGG))[i4];
    v.x *= rinv; v.y *= rinv; v.z *= rinv; v.w *= rinv;
    ((float4*)(ws + OFF_ENT_CUR))[i4] = v;
    float4 o = ((float4*)out)[i4];
    o.x += v.x; o.y += v.y; o.z += v.z; o.w += v.w;
    ((float4*)out)[i4] = o;
  } else if (i4 < NE4 + NU4) {
    long j = i4 - NE4;
    float rinv = 1.0f/sqrtf(ws[OFF_NORM+1]);
    float4 v = ((float4*)(ws + OFF_USR_AGG))[j];
    v.x *= rinv; v.y *= rinv; v.z *= rinv; v.w *= rinv;
    ((float4*)(ws + OFF_USR_CUR))[j] = v;
    float4 o = ((float4*)out)[NE4 + j];
    o.x += v.x; o.y += v.y; o.z += v.z; o.w += v.w;
    ((float4*)out)[NE4 + j] = o;
  }
}

// ---------------------------------------------------------------
extern "C" void kernel_launch(void* const* d_in, const int* in_sizes, int n_in,
                              void* d_out, int out_size, void* d_ws, size_t ws_size,
                              hipStream_t stream) {
  const float* user_emb   = (const float*)d_in[0];
  const float* entity_emb = (const float*)d_in[1];
  const float* latent     = (const float*)d_in[2];
  const float* weight     = (const float*)d_in[3];
  const float* att        = (const float*)d_in[4];
  const float* vals       = (const float*)d_in[5];
  const int*   eh         = (const int*)d_in[6];
  const int*   et         = (const int*)d_in[7];
  const int*   ety        = (const int*)d_in[8];
  const int*   ui         = (const int*)d_in[9];
  const int*   ei         = (const int*)d_in[10];
  float* out = (float*)d_out;
  float* ws  = (float*)d_ws;

  // cor scalar lives at the tail of d_out
  float* cor_out = out + (long)N_ENT*CH + (long)N_USERS*CH;

  k_small<<<1, 64, 0, stream>>>(att, weight, ws + OFF_DW, cor_out);
  k_init <<<18750, 256, 0, stream>>>(user_emb, entity_emb, ws, out);

  for (int hop = 0; hop < NHOPS; ++hop) {
    k_zero     <<<18946, 256, 0, stream>>>(ws);
    k_edge     <<<31250, 256, 0, stream>>>(ws + OFF_ENT_CUR, weight, eh, et, ety,
                                           ws + OFF_ENT_AGG, ws + OFF_CNT);
    k_nnz      <<<15625, 256, 0, stream>>>(ws + OFF_ENT_CUR, vals, ui, ei,
                                           ws + OFF_USR_AGG);
    k_entdiv   <<<12500, 256, 0, stream>>>(ws + OFF_ENT_AGG, ws + OFF_CNT,
                                           ws + OFF_NORM);
    k_userscore<<< 6250,  32, 0, stream>>>(ws + OFF_USR_CUR, latent, ws + OFF_DW,
                                           ws + OFF_USR_AGG, ws + OFF_NORM);
    k_finalize <<<18750, 256, 0, stream>>>(ws, out);
  }
}